// ObjectClassifierBlock_29180007809242
// MI455X (gfx1250) — compile-verified
//
#include <hip/hip_runtime.h>
#include <cstdint>
#include <cstddef>

typedef float v2f __attribute__((ext_vector_type(2)));
typedef float v8f __attribute__((ext_vector_type(8)));

// D = A(16x4,f32) * B(4x16,f32) + C(16x16,f32)  -- CDNA5 V_WMMA_F32_16X16X4_F32
__device__ __forceinline__ v8f wmma4(v2f a, v2f b, v8f c) {
  return __builtin_amdgcn_wmma_f32_16x16x4_f32(false, a, false, b, (short)0, c,
                                               false, false);
}

// CDNA5 async global->LDS copy (ASYNCcnt-tracked). Each lane supplies its own
// 16B global source and LDS destination; LDS address = low 32 bits of the
// generic pointer (ISA 10.2, LDS aperture).
__device__ __forceinline__ void async_ld_b128(uint32_t lds_addr,
                                              const void* gaddr) {
  asm volatile("global_load_async_to_lds_b128 %0, %1, off" ::"v"(lds_addr),
               "v"((uint64_t)(uintptr_t)gaddr)
               : "memory");
}
__device__ __forceinline__ void wait_async0() {
  asm volatile("s_wait_asynccnt 0x0" ::: "memory");
}

__device__ __forceinline__ float sigmoid_fast(float x) {
  return 1.0f / (1.0f + __expf(-x));
}
__device__ __forceinline__ float tanh_fast(float x) {
  x = fminf(fmaxf(x, -15.0f), 15.0f);
  float e = __expf(2.0f * x);
  return (e - 1.0f) / (e + 1.0f);
}

// Gate-axis permutation (N = 4H): permuted col n' = jg*64 + gate*16 + l
// maps to math col orig = gate*H + jg*16 + l.  Applied to B columns / bias in
// the GEMM and to U columns in the scan, so each wave's 4 accumulator tiles
// are exactly the i,f,g,o gates of one 16-wide h-group -> gate math stays in
// registers (no z spill through LDS on the serial scan path).
__device__ __forceinline__ int perm_to_orig(int np, int H) {
  return ((np >> 4) & 3) * H + ((np >> 6) << 4) + (np & 15);
}

// ---------------------------------------------------------------------------
// C[(b,t),n'] = sum_k A[m,k]*B[k,orig(n')] + bias[orig(n')],  m = b*T + t
// stored permuted over both the (b,t) axes -> [T,B,N] and the gate axis n'.
// Block: 256 thr (8 waves), tile M=128 x N=128, waves 4(M) x 2(N),
// per-wave 32x64 = 8 f32-WMMA accumulators. A & B staged in LDS via
// GLOBAL_LOAD_ASYNC_TO_LDS_B128 (no VGPR round-trip).
// ---------------------------------------------------------------------------
__global__ __launch_bounds__(256) void gemm_wmma_f32(
    const float* __restrict__ A, const float* __restrict__ Bm,
    const float* __restrict__ bias, float* __restrict__ C,
    int M, int K, int N, int T, long long sB, long long sT) {
  extern __shared__ float sm[];
  // LDS row stride for A: force (ldaS/4) odd so 16 row-strided lanes hit
  // 16 distinct banks (K=300 -> 300, K=128 -> 132).
  const int ldaS = ((K >> 2) & 1) ? K : K + 4;
  float* As = sm;
  float* Bs = sm + 128 * ldaS;

  const int tid = threadIdx.x;
  const int m0 = blockIdx.x * 128;
  const int H4 = N >> 2;  // = H of the consuming LSTM layer

  // stage A tile 128 x K async (rows are contiguous: A is packed [M,K])
  {
    const char* src = (const char*)(A + (size_t)m0 * (size_t)K);
    const uint32_t dstb = (uint32_t)(uintptr_t)As;
    const int k4 = K >> 2, lda4 = ldaS >> 2;
    for (int i = tid; i < 128 * k4; i += 256) {
      int r = i / k4, c = i - r * k4;
      async_ld_b128(dstb + (uint32_t)(r * lda4 + c) * 16u,
                    src + (size_t)i * 16);
    }
  }

  const int lane = tid & 31, wave = tid >> 5;
  const int half = lane >> 4, l15 = lane & 15;
  const int r0 = (wave >> 1) * 32;  // wave row offset within 128-row tile
  const int c0w = (wave & 1) * 64;  // wave col offset within 128-col subtile

  const int ntiles = N / 128;
  for (int nt = 0; nt < ntiles; ++nt) {
    // stage B tile K x 128 async with gate-axis permutation (the permutation
    // is identity within each 16-col group, so 16B chunks stay contiguous)
    {
      const char* src = (const char*)Bm;
      const uint32_t dstb = (uint32_t)(uintptr_t)Bs;
      const int n4 = N >> 2;
      for (int i = tid; i < K * 32; i += 256) {
        int r = i >> 5, c4 = i & 31;
        int np = nt * 128 + (c4 << 2);
        int orig = perm_to_orig(np, H4);
        async_ld_b128(dstb + (uint32_t)i * 16u,
                      src + ((size_t)r * n4 + (orig >> 2)) * 16);
      }
    }
    wait_async0();   // covers the A-tile loads too on the first iteration
    __syncthreads();

    v8f zero = {0.f, 0.f, 0.f, 0.f, 0.f, 0.f, 0.f, 0.f};
    v8f acc[2][4];
#pragma unroll
    for (int mi = 0; mi < 2; ++mi)
#pragma unroll
      for (int ni = 0; ni < 4; ++ni) acc[mi][ni] = zero;

    for (int k0 = 0; k0 < K; k0 += 4) {
      const int ka = k0 + 2 * half;  // lanes 0-15: K=k0,k0+1; 16-31: k0+2,k0+3
      v2f a[2], b[4];
#pragma unroll
      for (int mi = 0; mi < 2; ++mi) {
        const float* p = &As[(r0 + mi * 16 + l15) * ldaS + ka];
        a[mi].x = p[0];
        a[mi].y = p[1];
      }
#pragma unroll
      for (int ni = 0; ni < 4; ++ni) {
        const int c = c0w + ni * 16 + l15;
        b[ni].x = Bs[(ka << 7) + c];
        b[ni].y = Bs[((ka + 1) << 7) + c];
      }
#pragma unroll
      for (int mi = 0; mi < 2; ++mi)
#pragma unroll
        for (int ni = 0; ni < 4; ++ni)
          acc[mi][ni] = wmma4(a[mi], b[ni], acc[mi][ni]);
    }

    // store (+bias at the permuted source column), permuting m -> (b,t)
    float bv[4];
#pragma unroll
    for (int ni = 0; ni < 4; ++ni) {
      int np = nt * 128 + c0w + ni * 16 + l15;
      bv[ni] = bias[perm_to_orig(np, H4)];
    }
#pragma unroll
    for (int mi = 0; mi < 2; ++mi) {
#pragma unroll
      for (int r = 0; r < 8; ++r) {
        int m = m0 + r0 + mi * 16 + (half ? r + 8 : r);
        int bI = m / T, tI = m - bI * T;
        float* crow = C + (long long)bI * sB + (long long)tI * sT;
#pragma unroll
        for (int ni = 0; ni < 4; ++ni) {
          int np = nt * 128 + c0w + ni * 16 + l15;
          crow[np] = acc[mi][ni][r] + bv[ni];
        }
      }
    }
    __syncthreads();
  }
}

// ---------------------------------------------------------------------------
// LSTM scan. One block = ROWS batch rows across the whole sequence.
// U [H,4H] (gate-permuted) resident in LDS -- 256 KB for layer 1, only
// possible with CDNA5's 320 KB/WGP -- staged via async global->LDS.
// h in padded LDS rows, c in registers. Wave (rg,jg) owns rows [rg*16,+16)
// x h-cols [jg*16,+16) and computes the FOUR gate tiles for that patch, so
// i,f,g,o for each (row,j) live in one thread's 4 accumulators: gate math is
// pure register work, no z spill. xz_{t+1} is prefetched into registers
// while step t's WMMA chain runs; global_prefetch pulls t+2 toward the WGP.
// ---------------------------------------------------------------------------
template <int H, int ROWS>
__global__ __launch_bounds__(256) void lstm_scan_wmma(
    const float* __restrict__ xz,  // [T, Bsz, 4H]  gate-permuted cols
    const float* __restrict__ U,   // [H, 4H]       math layout (permuted here)
    float* __restrict__ hseq,      // [Bsz, T, H] or nullptr (math layout)
    float* __restrict__ hlast,     // [Bsz, H] or nullptr
    int Bsz, int T) {
  constexpr int FH = 4 * H;
  constexpr int JG = H / 16;  // h-col groups (waves along j)
  constexpr int HS = H + 4;   // padded h row: (HS/4) odd -> conflict-free
  static_assert(ROWS == 16 * (8 / JG), "block shape mismatch");

  extern __shared__ float sm[];
  float* Us = sm;           // [H][FH] permuted cols
  float* hs = Us + H * FH;  // [ROWS][HS]

  const int tid = threadIdx.x;
  const int b0 = blockIdx.x * ROWS;

  // one-time: U -> LDS (async, gate-axis permuted in 16B chunks), h = 0
  {
    const char* src = (const char*)U;
    const uint32_t dstb = (uint32_t)(uintptr_t)Us;
    constexpr int FH4 = FH / 4;
    for (int i = tid; i < H * FH4; i += 256) {
      int k = i / FH4, np = (i - k * FH4) << 2;
      int orig = perm_to_orig(np, H);
      async_ld_b128(dstb + (uint32_t)i * 16u,
                    src + ((size_t)k * FH + orig) * 4);
    }
  }
  for (int i = tid; i < ROWS * HS; i += 256) hs[i] = 0.0f;

  const int lane = tid & 31, wave = tid >> 5;
  const int half = lane >> 4, l15 = lane & 15;
  const int rg = wave / JG, jg = wave - rg * JG;
  const int rb = rg * 16;  // row-group base within block
  const int n0 = jg * 64;  // permuted col base: 4 gate tiles of 16

  float creg[8];
#pragma unroll
  for (int r = 0; r < 8; ++r) creg[r] = 0.0f;
  wait_async0();
  __syncthreads();

  v8f zero = {0.f, 0.f, 0.f, 0.f, 0.f, 0.f, 0.f, 0.f};

  // seed tiles for t = 0
  v8f cur[4];
#pragma unroll
  for (int g = 0; g < 4; ++g) {
#pragma unroll
    for (int r = 0; r < 8; ++r) {
      int rr = half ? r + 8 : r;
      cur[g][r] = xz[((size_t)0 * Bsz + b0 + rb + rr) * FH + n0 + g * 16 + l15];
    }
  }

  for (int t = 0; t < T; ++t) {
    // register-prefetch xz_{t+1}; L2 prefetch for t+2
    v8f nxt[4];
#pragma unroll
    for (int g = 0; g < 4; ++g) nxt[g] = zero;
    if (t + 1 < T) {
#pragma unroll
      for (int g = 0; g < 4; ++g) {
#pragma unroll
        for (int r = 0; r < 8; ++r) {
          int rr = half ? r + 8 : r;
          nxt[g][r] = xz[((size_t)(t + 1) * Bsz + b0 + rb + rr) * FH + n0 +
                         g * 16 + l15];
        }
      }
    }
    if (t + 2 < T) {
      // one 128B line per lane covers this wave's whole t+2 seed slab
      const float* pf =
          &xz[((size_t)(t + 2) * Bsz + b0 + rb + l15) * FH + n0 + half * 32];
      __builtin_prefetch(pf, 0, 2);
    }

    // z = xz_t + h @ U  (f32 WMMA, K in steps of 4) into cur
#pragma unroll
    for (int k0 = 0; k0 < H; k0 += 4) {
      const int ka = k0 + 2 * half;
      v2f a;
      a.x = hs[(rb + l15) * HS + ka];
      a.y = hs[(rb + l15) * HS + ka + 1];
#pragma unroll
      for (int g = 0; g < 4; ++g) {
        const int c = n0 + g * 16 + l15;
        v2f b;
        b.x = Us[ka * FH + c];
        b.y = Us[(ka + 1) * FH + c];
        cur[g] = wmma4(a, b, cur[g]);
      }
    }
    __syncthreads();  // all waves done reading hs

    // gates in registers: cur[0..3] = i,f,g,o for (row rr, col jg*16+l15)
#pragma unroll
    for (int r = 0; r < 8; ++r) {
      int rr = half ? r + 8 : r;
      float ig = sigmoid_fast(cur[0][r]);
      float fg = sigmoid_fast(cur[1][r]);
      float gg = tanh_fast(cur[2][r]);
      float og = sigmoid_fast(cur[3][r]);
      float c = fg * creg[r] + ig * gg;
      creg[r] = c;
      float h = og * tanh_fast(c);
      hs[(rb + rr) * HS + jg * 16 + l15] = h;
      if (hseq) hseq[((size_t)(b0 + rb + rr) * T + t) * H + jg * 16 + l15] = h;
    }
    __syncthreads();  // hs complete before next step's A-fragments

#pragma unroll
    for (int g = 0; g < 4; ++g) cur[g] = nxt[g];
  }

  if (hlast) {
    for (int i = tid; i < ROWS * H; i += 256) {
      int row = i / H, j = i & (H - 1);
      hlast[(b0 + row) * H + j] = hs[row * HS + j];
    }
  }
}

// ---------------------------------------------------------------------------
// out = softmax(h2_last @ Wd + bd)  -- tiny (0.46 MFLOP), one thread per row
// ---------------------------------------------------------------------------
__global__ void dense_softmax_kernel(const float* __restrict__ h2,
                                     const float* __restrict__ Wd,
                                     const float* __restrict__ bd,
                                     float* __restrict__ out, int Bsz) {
  int row = blockIdx.x * blockDim.x + threadIdx.x;
  if (row >= Bsz) return;
  float h[32];
#pragma unroll
  for (int j = 0; j < 32; ++j) h[j] = h2[row * 32 + j];
  float lg[14];
  float mx = -1e30f;
#pragma unroll
  for (int c = 0; c < 14; ++c) {
    float s = bd[c];
#pragma unroll
    for (int j = 0; j < 32; ++j) s += h[j] * Wd[j * 14 + c];
    lg[c] = s;
    mx = fmaxf(mx, s);
  }
  float se = 0.f;
#pragma unroll
  for (int c = 0; c < 14; ++c) {
    lg[c] = __expf(lg[c] - mx);
    se += lg[c];
  }
  float inv = 1.0f / se;
#pragma unroll
  for (int c = 0; c < 14; ++c) out[row * 14 + c] = lg[c] * inv;
}

extern "C" void kernel_launch(void* const* d_in, const int* in_sizes, int n_in,
                              void* d_out, int out_size, void* d_ws,
                              size_t ws_size, hipStream_t stream) {
  (void)in_sizes; (void)n_in; (void)out_size; (void)ws_size;
  const float* x  = (const float*)d_in[0];
  const float* W1 = (const float*)d_in[1];
  const float* U1 = (const float*)d_in[2];
  const float* b1 = (const float*)d_in[3];
  const float* W2 = (const float*)d_in[4];
  const float* U2 = (const float*)d_in[5];
  const float* b2 = (const float*)d_in[6];
  const float* Wd = (const float*)d_in[7];
  const float* bd = (const float*)d_in[8];
  float* out = (float*)d_out;

  constexpr int B = 512, T = 512, D = 300, H1 = 128, H2 = 32;
  constexpr int M = B * T;

  // workspace: xz region (reused by layer 2), h1 sequence, h2_last
  char* ws = (char*)d_ws;
  const size_t xz1_bytes = (size_t)T * B * 4 * H1 * sizeof(float);  // 512 MB
  const size_t h1_bytes  = (size_t)B * T * H1 * sizeof(float);      // 128 MB
  float* xz  = (float*)ws;                          // [T,B,4H1] then [T,B,4H2]
  float* h1  = (float*)(ws + xz1_bytes);            // [B,T,H1]
  float* h2l = (float*)(ws + xz1_bytes + h1_bytes); // [B,H2]

  const size_t gemm1_smem = (size_t)(128 * 300 + 300 * 128) * 4;    // 307200
  const size_t gemm2_smem = (size_t)(128 * 132 + 128 * 128) * 4;    // 133120
  const size_t lstm1_smem =
      (size_t)(H1 * 4 * H1 + 16 * (H1 + 4)) * 4;                    // 270592
  const size_t lstm2_smem =
      (size_t)(H2 * 4 * H2 + 64 * (H2 + 4)) * 4;                    // 25600

  hipFuncSetAttribute(reinterpret_cast<const void*>(&gemm_wmma_f32),
                      hipFuncAttributeMaxDynamicSharedMemorySize,
                      (int)gemm1_smem);
  hipFuncSetAttribute(reinterpret_cast<const void*>(&lstm_scan_wmma<H1, 16>),
                      hipFuncAttributeMaxDynamicSharedMemorySize,
                      (int)lstm1_smem);

  // 1) xz1 = x @ W1 + b1  -> [T,B,512], gate-permuted cols
  gemm_wmma_f32<<<M / 128, 256, gemm1_smem, stream>>>(
      x, W1, b1, xz, M, D, 4 * H1, T,
      (long long)(4 * H1), (long long)B * (4 * H1));
  // 2) layer-1 scan: 32 blocks x 16 rows -> h1 [B,T,128]
  lstm_scan_wmma<H1, 16><<<B / 16, 256, lstm1_smem, stream>>>(xz, U1, h1,
                                                              nullptr, B, T);
  // 3) xz2 = h1 @ W2 + b2 -> [T,B,128], gate-permuted (reuses xz region)
  gemm_wmma_f32<<<M / 128, 256, gemm2_smem, stream>>>(
      h1, W2, b2, xz, M, H1, 4 * H2, T,
      (long long)(4 * H2), (long long)B * (4 * H2));
  // 4) layer-2 scan: 8 blocks x 64 rows -> h2_last [B,32]
  lstm_scan_wmma<H2, 64><<<B / 64, 256, lstm2_smem, stream>>>(xz, U2, nullptr,
                                                              h2l, B, T);
  // 5) dense + softmax
  dense_softmax_kernel<<<(B + 255) / 256, 256, 0, stream>>>(h2l, Wd, bd, out,
                                                            B);
}